// Post_Prob_87196426043481
// MI455X (gfx1250) — compile-verified
//
#include <hip/hip_runtime.h>

typedef __attribute__((ext_vector_type(2))) float v2f;
typedef __attribute__((ext_vector_type(8))) float v8f;

namespace {
constexpr int NPTS = 512;
constexpr int CDIM = 64;
constexpr int MPIX = CDIM * CDIM;   // 4096 pixels
constexpr int ROWS = NPTS + 1;      // 513 softmax rows
constexpr float BG_RATIO = 0.15f;
constexpr float EPS = 1e-5f;
// logits = -dis / (2*SIGMA^2) = -dis/128 ;  exp(t) = exp2(t/ln2)
constexpr float K2 = 1.0f / (128.0f * 0.69314718055994531f);
}

__global__ __launch_bounds__(256)
void post_prob_kernel(const float* __restrict__ points,
                      const float* __restrict__ st_sizes,
                      float* __restrict__ out) {
  // Per-lane packed WMMA A-operands, in the exact 16x4 f32 A layout:
  //   sA[p]        = {1,  x_p}            (lanes 0-15 : K=0,1)
  //   sA[512 + p]  = {y_p, x_p^2+y_p^2}   (lanes 16-31: K=2,3)
  __shared__ v2f sA[2 * NPTS];  // 8 KB

  const int b   = blockIdx.y;
  const int tid = threadIdx.x;

  const float* pb = points + (size_t)b * NPTS * 2;
  for (int p = tid; p < NPTS; p += 256) {
    float x = pb[2 * p + 0];
    float y = pb[2 * p + 1];
    v2f lo, hi2;
    lo[0] = 1.0f; lo[1] = x;
    hi2[0] = y;   hi2[1] = __builtin_fmaf(x, x, y * y);
    sA[p]        = lo;
    sA[NPTS + p] = hi2;
  }
  __syncthreads();

  const int lane = tid & 31;
  const int wave = tid >> 5;
  const int m0   = blockIdx.x * 256 + wave * 32;  // 32 pixels per wave (two 16-col tiles)
  const int col  = lane & 15;
  const bool hi  = lane >= 16;

  // Single per-lane LDS base: one unconditional ds_load_b64 per chunk.
  const v2f* pa = sA + (hi ? NPTS : 0) + col;

  // B tile (4x16 f32): per-lane K-row = vgpr + 2*(lane>=16).
  // Rows: [cx^2+cy^2, -2cx, -2cy, 1]  => D = (x-cx)^2 + (y-cy)^2
  v2f bmat[2];
#pragma unroll
  for (int t = 0; t < 2; ++t) {
    int m = m0 + t * 16 + col;
    float cx = (float)((m & (CDIM - 1)) * 8 + 4);
    float cy = (float)((m >> 6) * 8 + 4);
    bmat[t][0] = hi ? (-2.0f * cy) : __builtin_fmaf(cx, cx, cy * cy);
    bmat[t][1] = hi ? 1.0f : (-2.0f * cx);
  }

  float bgS = st_sizes[b] * BG_RATIO;
  bgS *= bgS;

  const v8f zero = {};

  // ---------- pass 1: per-pixel min distance over all 512 points ----------
  float mn0 = 3.4e38f, mn1 = 3.4e38f;
  for (int ch = 0; ch < 32; ++ch) {
    v2f a = pa[ch * 16];
    v8f d0 = __builtin_amdgcn_wmma_f32_16x16x4_f32(false, a, false, bmat[0], (short)0, zero, false, false);
    v8f d1 = __builtin_amdgcn_wmma_f32_16x16x4_f32(false, a, false, bmat[1], (short)0, zero, false, false);
#pragma unroll
    for (int v = 0; v < 8; ++v) {
      mn0 = fminf(mn0, d0[v]);
      mn1 = fminf(mn1, d1[v]);
    }
  }
  mn0 = fminf(mn0, __shfl_xor(mn0, 16, 32));
  mn1 = fminf(mn1, __shfl_xor(mn1, 16, 32));
  mn0 = fmaxf(mn0, 0.0f);
  mn1 = fmaxf(mn1, 0.0f);

  const float bg0 = bgS / (mn0 + EPS);
  const float bg1 = bgS / (mn1 + EPS);
  const float dm0 = fminf(mn0, bg0);   // -128 * (max logit)
  const float dm1 = fminf(mn1, bg1);

  // ---------- pass 2: softmax denominators ----------
  float s0 = 0.0f, s1 = 0.0f;
  for (int ch = 0; ch < 32; ++ch) {
    v2f a = pa[ch * 16];
    v8f d0 = __builtin_amdgcn_wmma_f32_16x16x4_f32(false, a, false, bmat[0], (short)0, zero, false, false);
    v8f d1 = __builtin_amdgcn_wmma_f32_16x16x4_f32(false, a, false, bmat[1], (short)0, zero, false, false);
#pragma unroll
    for (int v = 0; v < 8; ++v) {
      s0 += __builtin_amdgcn_exp2f((dm0 - d0[v]) * K2);
      s1 += __builtin_amdgcn_exp2f((dm1 - d1[v]) * K2);
    }
  }
  s0 += __shfl_xor(s0, 16, 32);
  s1 += __shfl_xor(s1, 16, 32);
  const float eb0 = __builtin_amdgcn_exp2f((dm0 - bg0) * K2);
  const float eb1 = __builtin_amdgcn_exp2f((dm1 - bg1) * K2);
  s0 += eb0;
  s1 += eb1;
  const float inv0 = __builtin_amdgcn_rcpf(s0);
  const float inv1 = __builtin_amdgcn_rcpf(s1);

  // ---------- pass 3: recompute distances, write normalized probabilities ----------
  // Cross-lane tile merge: after the xor-16 swap, lanes 0-15 carry tile0 cols and
  // lanes 16-31 carry tile1 cols of the SAME point-row, so each store is 32
  // consecutive floats (128B) of one [b, p, :] row.
  float* ob = out + (size_t)b * ROWS * MPIX + m0 + lane;
  for (int ch = 0; ch < 32; ++ch) {
    v2f a = pa[ch * 16];
    v8f d0 = __builtin_amdgcn_wmma_f32_16x16x4_f32(false, a, false, bmat[0], (short)0, zero, false, false);
    v8f d1 = __builtin_amdgcn_wmma_f32_16x16x4_f32(false, a, false, bmat[1], (short)0, zero, false, false);
#pragma unroll
    for (int v = 0; v < 8; ++v) {
      float e0 = __builtin_amdgcn_exp2f((dm0 - d0[v]) * K2) * inv0;
      float e1 = __builtin_amdgcn_exp2f((dm1 - d1[v]) * K2) * inv1;
      float x0 = __shfl_xor(e0, 16, 32);
      float x1 = __shfl_xor(e1, 16, 32);
      float wlo = hi ? x1 : e0;  // point row ch*16+v   : 32 consecutive pixels
      float whi = hi ? e1 : x0;  // point row ch*16+v+8 : 32 consecutive pixels
      ob[(size_t)(ch * 16 + v) * MPIX]     = wlo;
      ob[(size_t)(ch * 16 + v + 8) * MPIX] = whi;
    }
  }
  // background row (p = 512), one coalesced store per wave
  ob[(size_t)NPTS * MPIX] = hi ? (eb1 * inv1) : (eb0 * inv0);
}

extern "C" void kernel_launch(void* const* d_in, const int* in_sizes, int n_in,
                              void* d_out, int out_size, void* d_ws, size_t ws_size,
                              hipStream_t stream) {
  const float* points   = (const float*)d_in[0];  // [B, 512, 2] f32
  const float* st_sizes = (const float*)d_in[1];  // [B] f32
  float* out = (float*)d_out;                     // [B, 513, 4096] f32

  const int B = in_sizes[1];                      // batch from st_sizes length
  dim3 grid(MPIX / 256, B);                       // 16 pixel-blocks x B
  post_prob_kernel<<<grid, 256, 0, stream>>>(points, st_sizes, out);
}